// DeepgazeSpadeV2_35270271435559
// MI455X (gfx1250) — compile-verified
//
#include <hip/hip_runtime.h>
#include <hip/hip_bf16.h>

// Problem constants (from reference)
#define B_   2
#define C_   768
#define HP   18
#define WP   18
#define P_   (HP * WP)     // 324 patch positions
#define S_   256
#define HI   256
#define WI   256
#define NPIX (HI * WI)     // 65536

typedef float v8f __attribute__((ext_vector_type(8)));
typedef float v2f __attribute__((ext_vector_type(2)));

__device__ __forceinline__ int seg_at_patch(const int* __restrict__ segmap, int b, int p) {
    int h  = p / WP, w = p % WP;
    int ih = (h * HI) / HP;          // matches arange(Hp)*Hi//Hp
    int iw = (w * WI) / WP;
    int s  = segmap[b * NPIX + ih * WI + iw];
    return min(max(s, 0), S_ - 1);
}

// -------------------------------------------------------------------------
// Phase 1: segment sums as one-hot GEMM via V_WMMA_F32_16X16X4_F32 (exact f32)
//   sums[b*S + s][c] = sum over patches p with seg(p)==s of feats[b][c][p]
// Grid: B_ * 2 * (C_/16) blocks of 256 threads (8 waves).
// Wave w of s-group g owns the 16x16 tile (s_tile = g*8+w, c_tile).
// -------------------------------------------------------------------------
__global__ void seg_sum_wmma(const float* __restrict__ feats,
                             const int*   __restrict__ segmap,
                             float*       __restrict__ sums) {
    const int CT16 = C_ / 16;                 // 48
    int bx    = blockIdx.x;
    int b     = bx / (2 * CT16);
    int rem   = bx % (2 * CT16);
    int sgrp  = rem / CT16;                   // 0..1
    int ctile = rem % CT16;                   // 0..47
    int c_base = ctile * 16;

    __shared__ int segp[P_];
    int t = threadIdx.x;
    if (t < P_) segp[t] = seg_at_patch(segmap, b, t);
    __syncthreads();

    int wave   = t >> 5;
    int lane   = t & 31;
    int s_base = (sgrp * 8 + wave) * 16;
    int M      = lane & 15;                   // row (A) == col (B) index for this lane
    int koff   = (lane < 16) ? 0 : 2;         // 16x4 f32 A/B lane layout: K={0,1} / {2,3}

    v8f acc = {0.f, 0.f, 0.f, 0.f, 0.f, 0.f, 0.f, 0.f};
    const float* fcol = feats + (size_t)(b * C_ + c_base + M) * P_;

    for (int kb = 0; kb < P_; kb += 4) {      // 81 iterations, K=4 each
        v2f a, bv;
        a.x  = (segp[kb + koff]     == s_base + M) ? 1.0f : 0.0f;
        a.y  = (segp[kb + koff + 1] == s_base + M) ? 1.0f : 0.0f;
        bv.x = fcol[kb + koff];
        bv.y = fcol[kb + koff + 1];
        acc = __builtin_amdgcn_wmma_f32_16x16x4_f32(
            /*neg_a=*/false, a, /*neg_b=*/false, bv,
            /*c_mod=*/(short)0, acc, /*reuse_a=*/false, /*reuse_b=*/false);
    }

    // C/D layout: VGPR r -> M = r (lanes 0-15) or r+8 (lanes 16-31); N = lane&15
    int n     = lane & 15;
    int mhalf = (lane < 16) ? 0 : 8;
    #pragma unroll
    for (int r = 0; r < 8; ++r) {
        int s = s_base + mhalf + r;
        sums[(size_t)(b * S_ + s) * C_ + c_base + n] = acc[r];
    }
}

// -------------------------------------------------------------------------
// Phase 2: per-(batch,segment) patch counts via LDS histogram
// -------------------------------------------------------------------------
__global__ void seg_counts(const int* __restrict__ segmap, int* __restrict__ counts) {
    int b = blockIdx.x;
    __shared__ int h[S_];
    int t = threadIdx.x;                      // 512 threads
    if (t < S_) h[t] = 0;
    __syncthreads();
    if (t < P_) atomicAdd(&h[seg_at_patch(segmap, b, t)], 1);
    __syncthreads();
    if (t < S_) counts[b * S_ + t] = h[t];
}

// -------------------------------------------------------------------------
// Phase 3: seg_avg = sums / max(count, 1)
// -------------------------------------------------------------------------
__global__ void finalize_avg(const float* __restrict__ sums,
                             const int*   __restrict__ counts,
                             float*       __restrict__ seg_avg) {
    int i = blockIdx.x * blockDim.x + threadIdx.x;
    if (i >= B_ * S_ * C_) return;
    int cnt = counts[i / C_];
    seg_avg[i] = sums[i] / (float)max(cnt, 1);
}

// -------------------------------------------------------------------------
// Phase 4: paint — the bandwidth-critical kernel (~402 MB of stores).
// Block = (batch, 16-channel tile, 2048-pixel tile). 256 threads, 8 px each.
// LDS holds seg_avg[0..255][c0..c0+15] at stride 17 (bank-conflict-free
// gather). Stores are 128B-coalesced per wave, non-temporal (no reuse).
// -------------------------------------------------------------------------
#define CT   16
#define PAD  17
#define PPT  8
__global__ void paint(const float* __restrict__ seg_avg,
                      const int*   __restrict__ segmap,
                      float*       __restrict__ out) {
    const int CTILES = C_ / CT;               // 48
    const int PTILES = NPIX / (256 * PPT);    // 32
    int bx  = blockIdx.x;
    int b   = bx / (CTILES * PTILES);
    int rem = bx % (CTILES * PTILES);
    int ct  = rem / PTILES;
    int pt  = rem % PTILES;
    int c0    = ct * CT;
    int pbase = pt * (256 * PPT);

    __shared__ float tab[S_ * PAD];
    int t = threadIdx.x;                      // 256 threads; thread t stages segment t
    const float* srow = seg_avg + (size_t)(b * S_ + t) * C_ + c0;
    #pragma unroll
    for (int j = 0; j < CT; ++j) tab[t * PAD + j] = srow[j];
    __syncthreads();

    const int* smap = segmap + b * NPIX;
    float* ob = out + (size_t)b * C_ * NPIX + (size_t)c0 * NPIX;
    #pragma unroll
    for (int pp = 0; pp < PPT; ++pp) {
        int pix = pbase + pp * 256 + t;       // lanes -> consecutive wi: coalesced
        int seg = smap[pix];
        seg = min(max(seg, 0), S_ - 1);
        const float* trow = &tab[seg * PAD];
        #pragma unroll
        for (int j = 0; j < CT; ++j) {
            __builtin_nontemporal_store(trow[j], &ob[(size_t)j * NPIX + pix]);
        }
    }
}

extern "C" void kernel_launch(void* const* d_in, const int* in_sizes, int n_in,
                              void* d_out, int out_size, void* d_ws, size_t ws_size,
                              hipStream_t stream) {
    const float* feats  = (const float*)d_in[0];
    const int*   segmap = (const int*)d_in[1];
    // d_in[2] = num_total_segments (== 256, baked in)
    float* out = (float*)d_out;

    // Workspace: sums (1.5MB) | seg_avg (1.5MB) | counts (2KB)
    float* sums    = (float*)d_ws;
    float* seg_avg = sums + (size_t)B_ * S_ * C_;
    int*   counts  = (int*)(seg_avg + (size_t)B_ * S_ * C_);

    seg_sum_wmma<<<B_ * 2 * (C_ / 16), 256, 0, stream>>>(feats, segmap, sums);
    seg_counts  <<<B_, 512, 0, stream>>>(segmap, counts);
    finalize_avg<<<(B_ * S_ * C_ + 255) / 256, 256, 0, stream>>>(sums, counts, seg_avg);
    paint       <<<B_ * (C_ / CT) * (NPIX / (256 * PPT)), 256, 0, stream>>>(seg_avg, segmap, out);
}